// BP_RNetwork_42039139894161
// MI455X (gfx1250) — compile-verified
//
#include <hip/hip_runtime.h>
#include <math.h>
#include <stdint.h>

// ---------------- problem constants ----------------
#define BATCH 256
#define ISIZE 64
#define HDIM  512
#define ADIM  8
#define CLIPVAL 2.0f

// ---------------- vector types ----------------
typedef float v2f  __attribute__((ext_vector_type(2)));
typedef float v8f  __attribute__((ext_vector_type(8)));
typedef unsigned int u32x4 __attribute__((ext_vector_type(4)));
typedef int   i32x4 __attribute__((ext_vector_type(4)));
typedef int   i32x8 __attribute__((ext_vector_type(8)));

// =====================================================================
// Kernel A: pre[b,n] = inputs @ i2h_w^T + i2h_b + h0 @ w
// fp32 WMMA 16x16x4. One wave per 16x16 output tile.
// 512 tiles = 16 (M) x 32 (N); 128 blocks x 4 waves.
// =====================================================================
__global__ __launch_bounds__(128) void pre_gemm_wmma(
    const float* __restrict__ inputs,   // [256,64]
    const float* __restrict__ h0,       // [256,512]
    const float* __restrict__ i2h_w,    // [512,64]
    const float* __restrict__ i2h_b,    // [512]
    const float* __restrict__ w,        // [512,512]
    float* __restrict__ pre)            // [256,512]
{
    const int wave = threadIdx.x >> 5;
    const int lane = threadIdx.x & 31;
    const int tile = blockIdx.x * 4 + wave;   // 0..511
    const int m0 = (tile >> 5) * 16;          // batch tile
    const int n0 = (tile & 31) * 16;          // output-neuron tile
    const int lm = lane & 15;                 // M (for A) / N (for B) index
    const int hi = lane >> 4;                 // K-pair select: {0,1} or {2,3}

    v8f acc = {};

    // ---- segment 1: inputs[256x64] x i2h_w^T  (B[k][n] = i2h_w[n*64+k]) ----
    {
        const float* Arow = inputs + (m0 + lm) * ISIZE;
        const float* Brow = i2h_w  + (n0 + lm) * ISIZE;
        for (int kk = 0; kk < ISIZE; kk += 4) {
            const int k0 = kk + 2 * hi;
            v2f a, b;
            a.x = Arow[k0];     a.y = Arow[k0 + 1];
            b.x = Brow[k0];     b.y = Brow[k0 + 1];
            acc = __builtin_amdgcn_wmma_f32_16x16x4_f32(
                false, a, false, b, (short)0, acc, false, false);
        }
    }

    // ---- segment 2: h0[256x512] x w  (B[k][n] = w[k*512+n]) ----
    {
        const float* Arow = h0 + (m0 + lm) * HDIM;
        for (int kk = 0; kk < HDIM; kk += 4) {
            const int k0 = kk + 2 * hi;
            v2f a, b;
            a.x = Arow[k0];                     a.y = Arow[k0 + 1];
            b.x = w[(size_t)k0 * HDIM + n0 + lm];
            b.y = w[(size_t)(k0 + 1) * HDIM + n0 + lm];
            acc = __builtin_amdgcn_wmma_f32_16x16x4_f32(
                false, a, false, b, (short)0, acc, false, false);
        }
    }

    // ---- store C tile + bias: VGPR r -> row m0 + r + 8*hi, col n0 + lm ----
    const float bias = i2h_b[n0 + lm];
#pragma unroll
    for (int r = 0; r < 8; ++r) {
        const int row = m0 + r + 8 * hi;
        pre[(size_t)row * HDIM + n0 + lm] = acc[r] + bias;
    }
}

// =====================================================================
// TDM: load `rows` x 512-float tile (row stride 512 floats) into LDS.
// D# packed per CDNA5 ISA sec 8 (2-D tensor, data_size = 4B).
// 6-arg builtin form (clang-23 / therock-10.0 headers).
// =====================================================================
__device__ __forceinline__ void tdm_load_rows(const float* gsrc,
                                              unsigned lds_byte_off,
                                              int rows)
{
    const unsigned long long ga = (unsigned long long)(uintptr_t)gsrc;
    u32x4 g0;
    g0[0] = 1u;                                            // count=1, user mode
    g0[1] = lds_byte_off;                                  // lds_addr
    g0[2] = (unsigned)(ga & 0xFFFFFFFFu);                  // global_addr[31:0]
    g0[3] = (unsigned)((ga >> 32) & 0x1FFFFFFu)            // global_addr[56:32]
          | (2u << 30);                                    // type = image
    i32x8 g1;
    g1[0] = 2 << 16;            // workgroup_mask=0, data_size=2 (4 bytes)
    g1[1] = HDIM << 16;         // tensor_dim0[15:0] = 512
    g1[2] = HDIM << 16;         // tensor_dim0[31:16]=0 | tensor_dim1[15:0]=512
    g1[3] = HDIM << 16;         // tensor_dim1[31:16]=0 | tile_dim0 = 512
    g1[4] = rows & 0xFFFF;      // tile_dim1 = rows, tile_dim2 = 0
    g1[5] = HDIM;               // tensor_dim0_stride[31:0] = 512
    g1[6] = 0;                  // stride hi / dim1_stride lo
    g1[7] = 0;
    const i32x4 z4 = {0, 0, 0, 0};
    const i32x8 z8 = {0, 0, 0, 0, 0, 0, 0, 0};
    __builtin_amdgcn_tensor_load_to_lds(g0, g1, z4, z4, z8, 0);
}

// block-wide sum over 256 threads (all threads receive the result)
__device__ __forceinline__ float block_reduce_256(float v, float* red)
{
    const int tid = threadIdx.x;
    red[tid] = v;
    __syncthreads();
#pragma unroll
    for (int s = 128; s >= 1; s >>= 1) {
        if (tid < s) red[tid] += red[tid + s];
        __syncthreads();
    }
    const float r = red[0];
    __syncthreads();
    return r;
}

// =====================================================================
// Kernel B: one block per sample b. 256 threads, each owns 2 columns.
//  pass 1: rec[k] = sum_h h0[h]*alpha[h,k]*hebb[b,h,k]   (TDM-staged)
//  hact = tanh(pre + rec); 10 block reductions; eta
//  pass 2: hebb_new = clip(hebb + (eta*mfw+mfb)*hact[k]*h0[h])
// =====================================================================
#define CH  16                   // rows per TDM chunk
#define NCH (HDIM / CH)          // 32 chunks

__global__ __launch_bounds__(256) void fused_plastic_step(
    const float* __restrict__ h0g,      // [256,512]
    const float* __restrict__ hebb,     // [256,512,512]
    const float* __restrict__ alpha,    // [512,512]
    const float* __restrict__ h2mod_w,  // [512]
    const float* __restrict__ h2mod_b,  // [1]
    const float* __restrict__ mfw,      // [512] (modfanout_w)
    const float* __restrict__ mfb,      // [512] (modfanout_b)
    const float* __restrict__ h2o_w,    // [8,512]
    const float* __restrict__ h2o_b,    // [8]
    const float* __restrict__ h2v_w,    // [512]
    const float* __restrict__ h2v_b,    // [1]
    float* __restrict__ out_act,        // [256,8]
    float* __restrict__ out_val,        // [256]
    float* __restrict__ out_hact,       // [256,512]  (holds `pre` on entry)
    float* __restrict__ out_hebb)       // [256,512,512]
{
    __shared__ float h0s[HDIM];
    __shared__ float red[256];
    __shared__ float tile[2][CH * HDIM];   // 2 x 32KB double buffer

    const int b   = blockIdx.x;
    const int tid = threadIdx.x;
    const int k0  = tid * 2;
    const size_t hb = (size_t)b * HDIM * HDIM;

    // stage h0 row; grab pre-activation (written by kernel A)
    h0s[tid]       = h0g[(size_t)b * HDIM + tid];
    h0s[tid + 256] = h0g[(size_t)b * HDIM + tid + 256];
    const float pre0 = out_hact[(size_t)b * HDIM + k0];
    const float pre1 = out_hact[(size_t)b * HDIM + k0 + 1];
    __syncthreads();

    const unsigned toff0 = (unsigned)(uintptr_t)(void*)&tile[0][0];
    const unsigned toff1 = (unsigned)(uintptr_t)(void*)&tile[1][0];

    // ------------- pass 1: plastic recurrence reduction -------------
    if (tid < 32) tdm_load_rows(hebb + hb, toff0, CH);

    float acc0 = 0.f, acc1 = 0.f;
    for (int c = 0; c < NCH; ++c) {
        if (tid < 32) {
            if (c + 1 < NCH) {
                tdm_load_rows(hebb + hb + (size_t)(c + 1) * CH * HDIM,
                              ((c + 1) & 1) ? toff1 : toff0, CH);
                __builtin_amdgcn_s_wait_tensorcnt(1);  // chunk c ready
            } else {
                __builtin_amdgcn_s_wait_tensorcnt(0);
            }
        }
        __syncthreads();

        const float* t = &tile[c & 1][0];
        const int hbase = c * CH;
#pragma unroll
        for (int h = 0; h < CH; ++h) {
            const float  hv = h0s[hbase + h];
            const float2 al = *(const float2*)(alpha + (size_t)(hbase + h) * HDIM + k0);
            acc0 = fmaf(hv * al.x, t[h * HDIM + k0],     acc0);
            acc1 = fmaf(hv * al.y, t[h * HDIM + k0 + 1], acc1);
        }
        __syncthreads();   // buffer may be reused next iteration
    }

    const float ha0 = tanhf(pre0 + acc0);
    const float ha1 = tanhf(pre1 + acc1);
    out_hact[(size_t)b * HDIM + k0]     = ha0;
    out_hact[(size_t)b * HDIM + k0 + 1] = ha1;

    // ------------- 10 block-wide dot products over hactiv -------------
    const float modsum = block_reduce_256(
        ha0 * h2mod_w[k0] + ha1 * h2mod_w[k0 + 1], red);
    const float valsum = block_reduce_256(
        ha0 * h2v_w[k0] + ha1 * h2v_w[k0 + 1], red);
    float osum[ADIM];
#pragma unroll
    for (int a = 0; a < ADIM; ++a) {
        osum[a] = block_reduce_256(
            ha0 * h2o_w[a * HDIM + k0] + ha1 * h2o_w[a * HDIM + k0 + 1], red);
    }
    const float eta = tanhf(modsum + h2mod_b[0]);
    if (tid == 0) {
        out_val[b] = valsum + h2v_b[0];
#pragma unroll
        for (int a = 0; a < ADIM; ++a)
            out_act[b * ADIM + a] = osum[a] + h2o_b[a];
    }

    // ------------- pass 2: Hebbian update (hebb hot in L2) -------------
    const float coef0 = (eta * mfw[k0]     + mfb[k0])     * ha0;
    const float coef1 = (eta * mfw[k0 + 1] + mfb[k0 + 1]) * ha1;
    const float* src = hebb     + hb + k0;
    float*       dst = out_hebb + hb + k0;
#pragma unroll 8
    for (int h = 0; h < HDIM; ++h) {
        const float2 hv = *(const float2*)(src + (size_t)h * HDIM);
        const float  s  = h0s[h];
        float n0v = fmaf(coef0, s, hv.x);
        float n1v = fmaf(coef1, s, hv.y);
        n0v = fminf(fmaxf(n0v, -CLIPVAL), CLIPVAL);
        n1v = fminf(fmaxf(n1v, -CLIPVAL), CLIPVAL);
        float2 o; o.x = n0v; o.y = n1v;
        *(float2*)(dst + (size_t)h * HDIM) = o;
    }
}

// =====================================================================
extern "C" void kernel_launch(void* const* d_in, const int* in_sizes, int n_in,
                              void* d_out, int out_size, void* d_ws, size_t ws_size,
                              hipStream_t stream)
{
    const float* inputs   = (const float*)d_in[0];
    const float* h0       = (const float*)d_in[1];
    const float* hebb     = (const float*)d_in[2];
    const float* i2h_w    = (const float*)d_in[3];
    const float* i2h_b    = (const float*)d_in[4];
    const float* w        = (const float*)d_in[5];
    const float* alpha    = (const float*)d_in[6];
    const float* h2mod_w  = (const float*)d_in[7];
    const float* h2mod_b  = (const float*)d_in[8];
    const float* mfw      = (const float*)d_in[9];
    const float* mfb      = (const float*)d_in[10];
    const float* h2o_w    = (const float*)d_in[11];
    const float* h2o_b    = (const float*)d_in[12];
    const float* h2v_w    = (const float*)d_in[13];
    const float* h2v_b    = (const float*)d_in[14];

    float* out      = (float*)d_out;
    float* out_act  = out;                                  // [256,8]
    float* out_val  = out + BATCH * ADIM;                   // [256]
    float* out_hact = out + BATCH * ADIM + BATCH;           // [256,512]
    float* out_hebb = out_hact + (size_t)BATCH * HDIM;      // [256,512,512]

    // Kernel A stages `pre` into the hactiv output slot (kernel B overwrites
    // it with tanh(pre+rec) after consuming it).
    pre_gemm_wmma<<<128, 128, 0, stream>>>(inputs, h0, i2h_w, i2h_b, w, out_hact);

    fused_plastic_step<<<BATCH, 256, 0, stream>>>(
        h0, hebb, alpha, h2mod_w, h2mod_b, mfw, mfb,
        h2o_w, h2o_b, h2v_w, h2v_b,
        out_act, out_val, out_hact, out_hebb);
}